// HyperScatteringModule_22393959481940
// MI455X (gfx1250) — compile-verified
//
#include <hip/hip_runtime.h>
#include <hip/hip_bf16.h>

#define N_NODES 50000
#define N_EDGES 25000
#define NNZ     1600000
#define D       128
#define N_STEPS 16
#define WROWS   6
#define WCOLS   17            // N_STEPS + 1
#define OUT_ROW (WROWS*D*2)   // 1536 floats per output row

typedef __attribute__((ext_vector_type(2))) float v2f;
typedef __attribute__((ext_vector_type(8))) float v8f;

// ---------------------------------------------------------------- utilities
__global__ void zero_kernel(float* __restrict__ p, long n4) {  // n4 = floats/4
  long i = (long)blockIdx.x * blockDim.x + threadIdx.x;
  long stride = (long)gridDim.x * blockDim.x;
  float4 z = make_float4(0.f, 0.f, 0.f, 0.f);
  float4* p4 = (float4*)p;
  for (; i < n4; i += stride) p4[i] = z;
}

__global__ void deg_kernel(const int* __restrict__ idx, float* __restrict__ deg, int nnz) {
  int i = blockIdx.x * blockDim.x + threadIdx.x;
  if (i < nnz) atomicAdd(&deg[idx[i]], 1.0f);
}

__global__ void invert_kernel(float* __restrict__ deg, int n) {
  int i = blockIdx.x * blockDim.x + threadIdx.x;
  if (i < n) {
    float d = deg[i];
    deg[i] = (d > 0.0f) ? 1.0f / d : 0.0f;
  }
}

// ------------------------------------------------------------- segment sum
// One wave32 per nonzero: gather one 512B row (4 floats/lane), scale by the
// source inverse degree (fused normalization), scatter-add to the dst row.
__global__ void scatter_kernel(const float* __restrict__ src,
                               const float* __restrict__ inv_deg_src,
                               const int* __restrict__ gidx,
                               const int* __restrict__ sidx,
                               float* __restrict__ dst, int nnz) {
  int wave = (int)((blockIdx.x * (unsigned)blockDim.x + threadIdx.x) >> 5);
  int lane = threadIdx.x & 31;
  if (wave >= nnz) return;
  // stream the index arrays ahead of use (gfx1250 global_prefetch_b8)
  __builtin_prefetch(gidx + wave + 16384, 0, 0);
  __builtin_prefetch(sidx + wave + 16384, 0, 0);
  int g = gidx[wave];
  int s = sidx[wave];
  float sc = inv_deg_src[g];
  float4 v = ((const float4*)(src + (long)g * D))[lane];
  float* dp = dst + (long)s * D + lane * 4;
  atomicAdd(dp + 0, v.x * sc);
  atomicAdd(dp + 1, v.y * sc);
  atomicAdd(dp + 2, v.z * sc);
  atomicAdd(dp + 3, v.w * sc);
}

// ----------------------------------------------------- WMMA wavelet update
// Incremental einsum: wc[i, p] += W[i, tcol] * level[p], done as a 16x16x4
// f32 WMMA. A(16x4): column K=0 holds W[:, tcol] (rows 6..15 zero).
// B(4x16): row K=0 holds 16 consecutive level values for one node/edge row.
// C/D(16x16 f32): rows 0..5 live, staged inside d_out at the a=0 relu slots
//   wc[i] @ out[row*1536 + i*256 + f*2].
// Stores write full {wc, 0} float2 pairs so each 16-lane row store is one
// contiguous 128B line (finalize overwrites the second slot anyway).
// One block (8 waves) per node/edge row; wave g handles features 16g..16g+15.
// EXEC is all-1s at the WMMA (early exit below is wave-uniform).
__global__ void wmma_wc_kernel(const float* __restrict__ level,
                               const float* __restrict__ wavelet,
                               int tcol, float* __restrict__ wc, int init) {
  int lane = threadIdx.x & 31;
  int g = threadIdx.x >> 5;
  float wv[WROWS];
#pragma unroll
  for (int r = 0; r < WROWS; ++r) wv[r] = wavelet[r * WCOLS + tcol];
  bool anynz = false;
#pragma unroll
  for (int r = 0; r < WROWS; ++r) anynz = anynz || (wv[r] != 0.0f);
  if (!init && !anynz) return;   // uniform: whole wave exits together

  int row = blockIdx.x;
  int f0 = g * 16;
  bool lo = lane < 16;

  // A: 16x4 f32. lanes 0-15: VGPR0=K0, VGPR1=K1; lanes 16-31: K2,K3 (zero).
  v2f a;
  a.x = (lo && lane < WROWS) ? wavelet[lane * WCOLS + tcol] : 0.0f;
  a.y = 0.0f;
  // B: 4x16 f32. lanes 0-15 VGPR0 = row K=0, N=lane. Everything else zero.
  v2f b;
  b.x = lo ? level[(long)row * D + f0 + lane] : 0.0f;
  b.y = 0.0f;

  // C: 16x16 f32 (lanes 0-15 VGPR r <-> M=r, N=lane). Only rows with a
  // nonzero wavelet entry participate; others stay zero and are not stored.
  long base = (long)row * OUT_ROW + (long)(f0 + (lane & 15)) * 2;
  v8f c = {};
  if (!init && lo) {
#pragma unroll
    for (int r = 0; r < WROWS; ++r)
      if (wv[r] != 0.0f) c[r] = wc[base + r * (D * 2)];
  }

  v8f d = __builtin_amdgcn_wmma_f32_16x16x4_f32(
      /*neg_a=*/false, a, /*neg_b=*/false, b,
      /*c_mod=*/(short)0, c, /*reuse_a=*/false, /*reuse_b=*/false);

  if (lo) {
#pragma unroll
    for (int r = 0; r < WROWS; ++r)
      if (init || wv[r] != 0.0f)
        *(float2*)(wc + base + r * (D * 2)) = make_float2(d[r], 0.0f);
  }
}

// ----------------------------------------------------------------- finalize
// Expand staged wc (a=0 slots) into (relu(x), relu(-x)) pairs; one b64 store
// per pair (8B-aligned, contiguous -> full-line writes).
__global__ void finalize_kernel(float* __restrict__ out, long total) {
  long t = (long)blockIdx.x * blockDim.x + threadIdx.x;
  if (t >= total) return;
  long row = t / (WROWS * D);
  int r = (int)(t % (WROWS * D));
  int w = r >> 7;
  int f = r & (D - 1);
  long base = row * OUT_ROW + (long)w * (D * 2) + (long)f * 2;
  float v = out[base];
  *(float2*)(out + base) = make_float2(fmaxf(v, 0.0f), fmaxf(-v, 0.0f));
}

// --------------------------------------------------------------------- host
extern "C" void kernel_launch(void* const* d_in, const int* in_sizes, int n_in,
                              void* d_out, int out_size, void* d_ws, size_t ws_size,
                              hipStream_t stream) {
  const float* X        = (const float*)d_in[0];   // [N_NODES, D]
  const float* Y        = (const float*)d_in[1];   // [N_EDGES, D]
  const int*   node_idx = (const int*)d_in[2];     // [NNZ]
  const int*   edge_idx = (const int*)d_in[3];     // [NNZ]
  const float* wavelet  = (const float*)d_in[4];   // [6, 17]
  float* out = (float*)d_out;

  // workspace carve: ~64.3 MB total
  float* ws        = (float*)d_ws;
  float* inv_deg_v = ws;                                  // N_NODES
  float* inv_deg_e = inv_deg_v + N_NODES;                 // N_EDGES
  float* xbuf0     = inv_deg_e + N_EDGES;                 // N_NODES*D
  float* xbuf1     = xbuf0 + (long)N_NODES * D;           // N_NODES*D
  float* ef        = xbuf1 + (long)N_NODES * D;           // N_EDGES*D

  float* out_nodes = out;
  float* out_edges = out + (long)N_NODES * OUT_ROW;

  // degrees -> inverse degrees
  zero_kernel<<<64, 256, 0, stream>>>(inv_deg_v, (long)N_NODES / 4);
  zero_kernel<<<64, 256, 0, stream>>>(inv_deg_e, (long)N_EDGES / 4);
  deg_kernel<<<(NNZ + 255) / 256, 256, 0, stream>>>(node_idx, inv_deg_v, NNZ);
  deg_kernel<<<(NNZ + 255) / 256, 256, 0, stream>>>(edge_idx, inv_deg_e, NNZ);
  invert_kernel<<<(N_NODES + 255) / 256, 256, 0, stream>>>(inv_deg_v, N_NODES);
  invert_kernel<<<(N_EDGES + 255) / 256, 256, 0, stream>>>(inv_deg_e, N_EDGES);

  // wc init with wavelet column 0 applied to X / Y (overwrites all 6 planes)
  wmma_wc_kernel<<<N_NODES, 256, 0, stream>>>(X, wavelet, 0, out_nodes, 1);
  wmma_wc_kernel<<<N_EDGES, 256, 0, stream>>>(Y, wavelet, 0, out_edges, 1);

  const float* xc = X;
  float* bufs[2] = {xbuf0, xbuf1};
  int scatter_blocks = (NNZ + 7) / 8;   // 8 wave32 per 256-thread block

  for (int t = 1; t <= N_STEPS; ++t) {
    // v2e: edge_feat = segsum(X_norm[node_idx] -> edge_idx)
    zero_kernel<<<2048, 256, 0, stream>>>(ef, (long)N_EDGES * D / 4);
    scatter_kernel<<<scatter_blocks, 256, 0, stream>>>(
        xc, inv_deg_v, node_idx, edge_idx, ef, NNZ);
    // e2v: node_feat = segsum(edge_feat_norm[edge_idx] -> node_idx)
    float* xn = bufs[(t - 1) & 1];
    zero_kernel<<<4096, 256, 0, stream>>>(xn, (long)N_NODES * D / 4);
    scatter_kernel<<<scatter_blocks, 256, 0, stream>>>(
        ef, inv_deg_e, edge_idx, node_idx, xn, NNZ);
    // incremental wavelet einsum (WMMA); no-op for all-zero wavelet columns
    wmma_wc_kernel<<<N_NODES, 256, 0, stream>>>(xn, wavelet, t, out_nodes, 0);
    wmma_wc_kernel<<<N_EDGES, 256, 0, stream>>>(ef, wavelet, t, out_edges, 0);
    xc = xn;
  }

  long total = (long)(N_NODES + N_EDGES) * (WROWS * D);   // 57.6M relu pairs
  finalize_kernel<<<(int)((total + 255) / 256), 256, 0, stream>>>(out, total);
}